// RotatedRetinaNetLoss_31250182045724
// MI455X (gfx1250) — compile-verified
//
#include <hip/hip_runtime.h>
#include <hip/hip_bf16.h>
#include <math.h>

// Problem constants (match reference setup_inputs)
#define NB   2
#define NA   16384
#define NG   16
#define NC   15
#define BLK  256
#define NBLK (NA / BLK)   // 64 blocks per image -> 64 partials = one 16x4 f32 WMMA A-matrix

typedef __attribute__((ext_vector_type(2))) float fvec2;
typedef __attribute__((ext_vector_type(8))) float fvec8;

struct V2 { float x, y; };

__device__ __forceinline__ void make_corners(float cx, float cy, float w, float h,
                                             float tdeg, V2 p[4]) {
    const float DEG2RAD = 0.017453292519943295f;
    float r = tdeg * DEG2RAD;
    float c = cosf(r), s = sinf(r);
    float hw = 0.5f * w, hh = 0.5f * h;
    const float lx[4] = {-hw,  hw, hw, -hw};
    const float ly[4] = {-hh, -hh, hh,  hh};
#pragma unroll
    for (int i = 0; i < 4; ++i) {
        p[i].x = cx + lx[i] * c - ly[i] * s;
        p[i].y = cy + lx[i] * s + ly[i] * c;
    }
}

__device__ __forceinline__ bool inside_quad(V2 p, const V2 q[4]) {
    bool ok = true;
#pragma unroll
    for (int j = 0; j < 4; ++j) {
        V2 a = q[j], b = q[(j + 1) & 3];
        float cr = (b.x - a.x) * (p.y - a.y) - (b.y - a.y) * (p.x - a.x);
        ok &= (cr >= -1e-6f);
    }
    return ok;
}

// Exact intersection area of two convex CCW quads (mirrors reference _inter_area).
__device__ float inter_area(const V2 p1[4], const V2 p2[4]) {
    V2   pts[24];
    bool m[24];
#pragma unroll
    for (int i = 0; i < 4; ++i) { pts[i] = p1[i];     m[i]     = inside_quad(p1[i], p2); }
#pragma unroll
    for (int i = 0; i < 4; ++i) { pts[4 + i] = p2[i]; m[4 + i] = inside_quad(p2[i], p1); }
    int k = 8;
#pragma unroll
    for (int i = 0; i < 4; ++i) {
        V2 a = p1[i], b = p1[(i + 1) & 3];
        float rx = b.x - a.x, ry = b.y - a.y;
#pragma unroll
        for (int j = 0; j < 4; ++j) {
            V2 c = p2[j], d = p2[(j + 1) & 3];
            float ux = d.x - c.x, uy = d.y - c.y;
            float den = rx * uy - ry * ux;
            float qx = c.x - a.x, qy = c.y - a.y;
            float safe = (fabsf(den) > 1e-10f) ? den : 1.0f;
            float t = (qx * uy - qy * ux) / safe;
            float s = (qx * ry - qy * rx) / safe;
            bool mi = (fabsf(den) > 1e-10f) && (t >= 0.f) && (t <= 1.f) && (s >= 0.f) && (s <= 1.f);
            pts[k].x = a.x + t * rx;
            pts[k].y = a.y + t * ry;
            m[k] = mi;
            ++k;
        }
    }
    int n = 0;
    float cx = 0.f, cy = 0.f;
    for (int i = 0; i < 24; ++i)
        if (m[i]) { ++n; cx += pts[i].x; cy += pts[i].y; }
    float inv = 1.0f / fmaxf((float)n, 1.0f);
    cx *= inv; cy *= inv;
    V2 ref = pts[0];
    for (int i = 23; i >= 0; --i) if (m[i]) ref = pts[i];   // first valid point
    for (int i = 0; i < 24; ++i) if (!m[i]) pts[i] = ref;   // dupes add 0 area
    float ang[24];
    for (int i = 0; i < 24; ++i) ang[i] = atan2f(pts[i].y - cy, pts[i].x - cx);
    // stable insertion sort by angle (matches stable argsort)
    for (int i = 1; i < 24; ++i) {
        float a = ang[i]; V2 p = pts[i];
        int j = i - 1;
        while (j >= 0 && ang[j] > a) { ang[j + 1] = ang[j]; pts[j + 1] = pts[j]; --j; }
        ang[j + 1] = a; pts[j + 1] = p;
    }
    float area = 0.f;
    for (int i = 0; i < 24; ++i) {
        V2 p = pts[i], q = pts[(i + 1) % 24];
        area += p.x * q.y - q.x * p.y;
    }
    area = 0.5f * fabsf(area);
    return (n >= 3) ? area : 0.0f;
}

__device__ __forceinline__ float py_rem180(float x) {   // jnp.remainder(x, 180): sign of divisor
    float r = fmodf(x, 180.0f);
    return (r < 0.0f) ? r + 180.0f : r;
}
__device__ __forceinline__ float smooth_l1(float d) {   // BETA = 1
    return (d < 1.0f) ? 0.5f * d * d : d - 0.5f;
}
__device__ __forceinline__ float log_sigmoid(float x) {
    return fminf(x, 0.0f) - log1pf(expf(-fabsf(x)));
}

// ---------------------------------------------------------------- kernels ---

__global__ void init_kernel(unsigned long long* bestkey) {
    int t = threadIdx.x;
    if (t < NB * NG) bestkey[t] = 0ULL;
}

__global__ void assign_kernel(const float* __restrict__ anchors,
                              const float* __restrict__ gt_boxes,
                              const int*   __restrict__ gt_labels,
                              int* __restrict__ lab, int* __restrict__ midx,
                              unsigned long long* __restrict__ bestkey) {
    int b = blockIdx.y;
    int i = blockIdx.x * BLK + threadIdx.x;

    __shared__ float sg[NG * 5];
    __shared__ int   sl[NG];
    if (threadIdx.x < NG * 5) sg[threadIdx.x] = gt_boxes[b * NG * 5 + threadIdx.x];
    if (threadIdx.x < NG)     sl[threadIdx.x] = gt_labels[b * NG + threadIdx.x];
    __syncthreads();

    float ax = anchors[i * 5 + 0], ay = anchors[i * 5 + 1];
    float aw = anchors[i * 5 + 2], ah = anchors[i * 5 + 3];
    float at = anchors[i * 5 + 4];
    V2 pa[4];
    make_corners(ax, ay, aw, ah, at, pa);
    float aa   = aw * ah;
    float a_sz = fmaxf(aw, ah);

    float maxiou = 0.0f;
    int   idx    = 0;
    for (int g = 0; g < NG; ++g) {
        float gx = sg[g * 5 + 0], gy = sg[g * 5 + 1];
        float gw = sg[g * 5 + 2], gh = sg[g * 5 + 3];
        float gt = sg[g * 5 + 4];
        float dx = ax - gx, dy = ay - gy;
        float iou = 0.0f;
        if (sqrtf(dx * dx + dy * dy) < 0.7f * (a_sz + fmaxf(gw, gh))) {
            V2 pg[4];
            make_corners(gx, gy, gw, gh, gt, pg);
            float inter = inter_area(pa, pg);
            iou = inter / (aa + gw * gh - inter + 1e-8f);
        }
        if (iou > maxiou) { maxiou = iou; idx = g; }    // first-max argmax semantics
        if (iou > 0.0f) {
            // key: higher IoU wins; among ties, LOWEST anchor index wins (jnp.argmax)
            unsigned long long key =
                ((unsigned long long)__float_as_uint(iou) << 32) |
                (unsigned long long)(0xFFFFFFFFu - (unsigned)i);
            atomicMax(&bestkey[b * NG + g], key);
        }
    }
    int l, mi = -1;
    if (maxiou >= 0.5f)      { l = sl[idx] + 1; mi = idx; }
    else if (maxiou < 0.4f)  { l = 0; }
    else                     { l = -1; }
    lab[b * NA + i]  = l;
    midx[b * NA + i] = mi;
}

// Sequential low-quality force match, one thread per image (mirrors the Python loop).
__global__ void force_kernel(const int* __restrict__ gt_labels,
                             const unsigned long long* __restrict__ bestkey,
                             int* __restrict__ lab, int* __restrict__ midx) {
    int b = threadIdx.x;
    if (b >= NB) return;
    for (int j = 0; j < NG; ++j) {
        unsigned long long key = bestkey[b * NG + j];
        if (key != 0ULL) {  // has[j]: some anchor had iou > 0
            int bi  = (int)(0xFFFFFFFFu - (unsigned)(key & 0xFFFFFFFFULL));
            int lbl = gt_labels[b * NG + j] + 1;
            if (lab[b * NA + bi] != lbl) {
                lab[b * NA + bi]  = lbl;
                midx[b * NA + bi] = j;
            }
        }
    }
}

__global__ void loss_kernel(const float* __restrict__ logits,
                            const float* __restrict__ box_reg,
                            const float* __restrict__ anchors,
                            const float* __restrict__ gt_boxes,
                            const int*   __restrict__ lab,
                            const int*   __restrict__ midx,
                            float* __restrict__ partials) {
    int b = blockIdx.y;
    int i = blockIdx.x * BLK + threadIdx.x;
    int l = lab[b * NA + i];
    bool pos = (l > 0), valid = (l >= 0);

    const float* lg = logits + ((size_t)b * NA + i) * NC;
    __builtin_prefetch(lg, 0, 0);   // global_prefetch_b8

    // focal loss over 15 classes
    float fl = 0.0f;
#pragma unroll
    for (int c = 0; c < NC; ++c) {
        float x  = lg[c];
        float tt = (pos && (c == l - 1)) ? 1.0f : 0.0f;
        float p  = 1.0f / (1.0f + expf(-x));
        float p_t     = p * tt + (1.0f - p) * (1.0f - tt);
        float alpha_t = 0.25f * tt + 0.75f * (1.0f - tt);
        float bce = -(tt * log_sigmoid(x) + (1.0f - tt) * log_sigmoid(-x));
        float om  = 1.0f - p_t;
        fl += alpha_t * om * om * bce;       // GAMMA = 2
    }
    float flv = valid ? fl : 0.0f;

    // smooth-L1 regression on positives
    float reg = 0.0f;
    if (pos) {
        int mi = midx[b * NA + i];
        const float* g  = gt_boxes + (b * NG + mi) * 5;
        const float* br = box_reg + ((size_t)b * NA + i) * 5;
        float ax = anchors[i * 5 + 0], ay = anchors[i * 5 + 1];
        float aw = anchors[i * 5 + 2], ah = anchors[i * 5 + 3];
        float at = anchors[i * 5 + 4];
        float d0 = (g[0] - ax) / aw;
        float d1 = (g[1] - ay) / ah;
        float d2 = logf(g[2] / aw);
        float d3 = logf(g[3] / ah);
        float dth = py_rem180(g[4] - at + 90.0f) - 90.0f;
        float s = 0.0f;
        s += smooth_l1(fabsf(br[0] - d0));
        s += smooth_l1(fabsf(br[1] - d1));
        s += smooth_l1(fabsf(br[2] - d2));
        s += smooth_l1(fabsf(br[3] - d3));
        float da = fabsf(py_rem180(br[4] - dth + 90.0f) - 90.0f);
        s += smooth_l1(da);
        reg = s;                              // REG_WEIGHT = 1
    }

    // deterministic block tree reductions
    __shared__ float red[BLK];
    float vals[3] = {flv, pos ? 1.0f : 0.0f, reg};
#pragma unroll
    for (int q = 0; q < 3; ++q) {
        red[threadIdx.x] = vals[q];
        __syncthreads();
        for (int s = BLK / 2; s > 0; s >>= 1) {
            if (threadIdx.x < s) red[threadIdx.x] += red[threadIdx.x + s];
            __syncthreads();
        }
        if (threadIdx.x == 0)
            partials[(b * 3 + q) * NBLK + blockIdx.x] = red[0];
        __syncthreads();
    }
}

// One wave. Each of the 6 (image, quantity) groups has exactly 64 partials = a
// 16x4 f32 A-matrix; D = A x ones reduces all 64 to row-sums in a single
// V_WMMA_F32_16X16X4_F32; shfl_xor(16) finishes the sum.
__global__ void finalize_kernel(const float* __restrict__ partials, float* __restrict__ out) {
    int lane = threadIdx.x;   // 32 lanes, EXEC all ones at the WMMA
    float sums[NB][3];
#pragma unroll
    for (int b = 0; b < NB; ++b) {
#pragma unroll
        for (int q = 0; q < 3; ++q) {
            const float* p = partials + (b * 3 + q) * NBLK;
            fvec2 a;  a.x = p[2 * lane]; a.y = p[2 * lane + 1];
            fvec2 bb; bb.x = 1.0f;       bb.y = 1.0f;
            fvec8 c = {};
            c = __builtin_amdgcn_wmma_f32_16x16x4_f32(
                    false, a, false, bb, (short)0, c, false, false);
            float s = c[0] + c[1] + c[2] + c[3] + c[4] + c[5] + c[6] + c[7];
            s += __shfl_xor(s, 16, 32);   // combine rows 0-7 (lanes<16) with rows 8-15
            sums[b][q] = s;
        }
    }
    if (lane == 0) {
        float cls = 0.0f, reg = 0.0f;
#pragma unroll
        for (int b = 0; b < NB; ++b) {
            float npos = fmaxf(sums[b][1], 1.0f);
            cls += sums[b][0] / npos;
            reg += sums[b][2] / npos;
        }
        cls *= (1.0f / NB);
        reg *= (1.0f / NB);
        out[0] = cls + reg;
        out[1] = cls;
        out[2] = reg;
    }
}

// ---------------------------------------------------------------- launch ----

extern "C" void kernel_launch(void* const* d_in, const int* in_sizes, int n_in,
                              void* d_out, int out_size, void* d_ws, size_t ws_size,
                              hipStream_t stream) {
    (void)in_sizes; (void)n_in; (void)out_size; (void)ws_size;
    const float* cls_logits = (const float*)d_in[0];   // (B,N,C)
    const float* box_reg    = (const float*)d_in[1];   // (B,N,5)
    const float* anchors    = (const float*)d_in[2];   // (N,5)
    const float* gt_boxes   = (const float*)d_in[3];   // (B,G,5)
    const int*   gt_labels  = (const int*)d_in[4];     // (B,G)
    float* out = (float*)d_out;                        // [total, cls, reg]

    char* ws = (char*)d_ws;
    int* lab  = (int*)ws;                                    // NB*NA
    int* midx = lab + NB * NA;                               // NB*NA
    unsigned long long* bestkey = (unsigned long long*)(midx + NB * NA);  // NB*NG
    float* partials = (float*)(bestkey + NB * NG);           // NB*3*NBLK

    init_kernel<<<1, 64, 0, stream>>>(bestkey);
    assign_kernel<<<dim3(NBLK, NB), BLK, 0, stream>>>(anchors, gt_boxes, gt_labels,
                                                      lab, midx, bestkey);
    force_kernel<<<1, NB, 0, stream>>>(gt_labels, bestkey, lab, midx);
    loss_kernel<<<dim3(NBLK, NB), BLK, 0, stream>>>(cls_logits, box_reg, anchors,
                                                    gt_boxes, lab, midx, partials);
    finalize_kernel<<<1, 32, 0, stream>>>(partials, out);
}